// ResidualSchedulingGNN_21912923144306
// MI455X (gfx1250) — compile-verified
//
#include <hip/hip_runtime.h>
#include <math.h>

#define SIGMA_F 1.0f
#define BN_EPS_F 1e-5f

#define N_OP   200000
#define N_MACH 1000
#define N_JOB  20000
#define E_OM   1000000
#define E_JO   200000
#define N_PAIRS 500000

typedef float v2f __attribute__((ext_vector_type(2)));
typedef float v8f __attribute__((ext_vector_type(8)));

// ---------------------------------------------------------------- utilities
__global__ void zero_kernel(float* __restrict__ p, long n) {
  long i = (long)blockIdx.x * blockDim.x + threadIdx.x;
  if (i < n) p[i] = 0.0f;
}

// MultiPeriodicEncoder: out[i, 0:16] = sin(x@W / sigma), out[i,16:32] = cos(...)
__global__ void encode_kernel(const float* __restrict__ x, const float* __restrict__ W,
                              float* __restrict__ out, int N, int Din) {
  int i = blockIdx.x * blockDim.x + threadIdx.x;
  if (i >= N) return;
  float xv[8];
  for (int d = 0; d < Din; ++d) xv[d] = x[i * Din + d];
  for (int j = 0; j < 16; ++j) {
    float h = 0.0f;
    for (int d = 0; d < Din; ++d) h += xv[d] * W[d * 16 + j];
    h /= SIGMA_F;
    out[i * 32 + j]      = sinf(h);
    out[i * 32 + 16 + j] = cosf(h);
  }
}

// segment_sum: agg[dst[e], :] += src_feat[src[e], :]   (float4 per thread)
__global__ void scatter_add4_kernel(const float* __restrict__ src_feat,
                                    const int* __restrict__ src_idx,
                                    const int* __restrict__ dst_idx,
                                    float* __restrict__ agg, int E, int F) {
  long t = (long)blockIdx.x * blockDim.x + threadIdx.x;
  int q = F >> 2;
  if (t >= (long)E * q) return;
  int e = (int)(t / q), g = (int)(t % q);
  int s = src_idx[e], d = dst_idx[e];
  const float4 v = *(const float4*)(src_feat + (long)s * F + g * 4);
  float* dst = agg + (long)d * F + g * 4;
  atomicAdd(dst + 0, v.x);
  atomicAdd(dst + 1, v.y);
  atomicAdd(dst + 2, v.z);
  atomicAdd(dst + 3, v.w);
}

// stats layout: [0:64]=colsum  [64:128]=colsumsq  [128:192]=bnA  [192:256]=bnB
__global__ void bn_finalize_kernel(float* __restrict__ stats, const float* __restrict__ g,
                                   const float* __restrict__ be, float invN, int M) {
  int t = threadIdx.x;
  if (t >= M) return;
  float mean = stats[t] * invN;
  float var  = stats[64 + t] * invN - mean * mean;
  float a    = g[t] * rsqrtf(var + BN_EPS_F);
  stats[128 + t] = a;
  stats[192 + t] = be[t] - mean * a;
}

// ------------------------------------------------------------- WMMA kernels
// H1: Y[N,64] = ((1+eps)*X + AGG) @ W1[K,64] + b1 ; accumulate column sum/sumsq.
// Block = 8 waves; tile 32 rows x 64 cols. A & W staged in LDS (conflict-free
// padded strides), fused elementwise applied once during staging.
template <int K>
__global__ void __launch_bounds__(256)
gin_h1_wmma(const float* __restrict__ X, const float* __restrict__ AGG,
            const float* __restrict__ epsp,
            const float* __restrict__ W1, const float* __restrict__ b1,
            float* __restrict__ Y, float* __restrict__ stats, int N) {
  constexpr int AS = K + 4;              // padded stride (== 4 mod 32)
  constexpr int Q  = K / 4;              // float4 groups per row
  __shared__ float sA[32 * AS];          // fused A tile
  __shared__ float sW[64 * AS];          // W transposed: sW[col*AS + k]
  __shared__ float ssum[64], ssq[64];
  const int tid = threadIdx.x;
  const int lane = tid & 31, wave = tid >> 5;
  const int lmod = lane & 15, lhalf = lane >> 4;
  const int mblk = blockIdx.x * 32;
  const int m0 = mblk + (wave >> 2) * 16;
  const int n0 = (wave & 3) * 16;
  const float pe = 1.0f + epsp[0];

  if (tid < 64) { ssum[tid] = 0.0f; ssq[tid] = 0.0f; }
  // stage A: rows clamped (garbage rows masked at store/stats time)
  for (int idx = tid; idx < 32 * Q; idx += 256) {
    int r = idx / Q, k = (idx % Q) * 4;
    int row = mblk + r; row = row < N ? row : N - 1;
    long g = (long)row * K + k;
    float4 xv = *(const float4*)(X + g);
    float4 av = *(const float4*)(AGG + g);
    float* d = &sA[r * AS + k];
    d[0] = pe * xv.x + av.x; d[1] = pe * xv.y + av.y;
    d[2] = pe * xv.z + av.z; d[3] = pe * xv.w + av.w;
  }
  // stage W transposed (float4 global read, scalar LDS writes)
  for (int idx = tid; idx < K * 16; idx += 256) {
    int k = idx >> 4, c = (idx & 15) * 4;
    float4 w = *(const float4*)(W1 + k * 64 + c);
    sW[(c + 0) * AS + k] = w.x;
    sW[(c + 1) * AS + k] = w.y;
    sW[(c + 2) * AS + k] = w.z;
    sW[(c + 3) * AS + k] = w.w;
  }
  __syncthreads();

  const int ar = (wave >> 2) * 16 + lmod;
  const int col = n0 + lmod;
  v8f c = {};
#pragma unroll
  for (int kk = 0; kk < K; kk += 4) {
    const int kb = kk + 2 * lhalf;
    v2f a = *(const v2f*)&sA[ar * AS + kb];
    v2f b = *(const v2f*)&sW[col * AS + kb];
    c = __builtin_amdgcn_wmma_f32_16x16x4_f32(false, a, false, b, (short)0, c, false, false);
  }
  float bias = b1[col];
  float s = 0.0f, sq = 0.0f;
#pragma unroll
  for (int r = 0; r < 8; ++r) {
    int row = m0 + r + 8 * lhalf;
    if (row < N) {
      float v = c[r] + bias;
      Y[(long)row * 64 + col] = v;
      s += v; sq += v * v;
    }
  }
  atomicAdd(&ssum[col], s);
  atomicAdd(&ssq[col], sq);
  __syncthreads();
  if (tid < 64) {
    atomicAdd(&stats[tid], ssum[tid]);
    atomicAdd(&stats[64 + tid], ssq[tid]);
  }
}

// H2: Y[N,64] (op)= ReLU(bnA*H1+bnB) @ W2[64,64] + b2 (+resid) (+=Y if accum)
// BN+ReLU applied once during LDS staging.
__global__ void __launch_bounds__(256)
gin_h2_wmma(const float* __restrict__ H1, const float* __restrict__ stats,
            const float* __restrict__ W2, const float* __restrict__ b2,
            const float* __restrict__ resid, float* __restrict__ Y,
            int N, int accum) {
  constexpr int K = 64, AS = 68;
  __shared__ float sA[32 * AS];
  __shared__ float sW[64 * AS];
  const int tid = threadIdx.x;
  const int lane = tid & 31, wave = tid >> 5;
  const int lmod = lane & 15, lhalf = lane >> 4;
  const int mblk = blockIdx.x * 32;
  const int m0 = mblk + (wave >> 2) * 16;
  const int n0 = (wave & 3) * 16;
  const float* bnA = stats + 128;
  const float* bnB = stats + 192;

  for (int idx = tid; idx < 32 * 16; idx += 256) {
    int r = idx >> 4, k = (idx & 15) * 4;
    int row = mblk + r; row = row < N ? row : N - 1;
    float4 h = *(const float4*)(H1 + (long)row * 64 + k);
    float* d = &sA[r * AS + k];
    d[0] = fmaxf(bnA[k + 0] * h.x + bnB[k + 0], 0.0f);
    d[1] = fmaxf(bnA[k + 1] * h.y + bnB[k + 1], 0.0f);
    d[2] = fmaxf(bnA[k + 2] * h.z + bnB[k + 2], 0.0f);
    d[3] = fmaxf(bnA[k + 3] * h.w + bnB[k + 3], 0.0f);
  }
  for (int idx = tid; idx < K * 16; idx += 256) {
    int k = idx >> 4, c = (idx & 15) * 4;
    float4 w = *(const float4*)(W2 + k * 64 + c);
    sW[(c + 0) * AS + k] = w.x;
    sW[(c + 1) * AS + k] = w.y;
    sW[(c + 2) * AS + k] = w.z;
    sW[(c + 3) * AS + k] = w.w;
  }
  __syncthreads();

  const int ar = (wave >> 2) * 16 + lmod;
  const int col = n0 + lmod;
  v8f c = {};
#pragma unroll
  for (int kk = 0; kk < K; kk += 4) {
    const int kb = kk + 2 * lhalf;
    v2f a = *(const v2f*)&sA[ar * AS + kb];
    v2f b = *(const v2f*)&sW[col * AS + kb];
    c = __builtin_amdgcn_wmma_f32_16x16x4_f32(false, a, false, b, (short)0, c, false, false);
  }
  float bias = b2[col];
#pragma unroll
  for (int r = 0; r < 8; ++r) {
    int row = m0 + r + 8 * lhalf;
    if (row < N) {
      float v = c[r] + bias;
      if (resid) v += resid[(long)row * 64 + col];
      if (accum) v += Y[(long)row * 64 + col];
      Y[(long)row * 64 + col] = v;
    }
  }
}

// Scoring layer 1: gathered concat A (K=192) staged in LDS (dedups the
// scattered gather 4x); W read from global (L1-resident); +BN stats.
__global__ void __launch_bounds__(256)
score_h1_wmma(const float* __restrict__ FO, const float* __restrict__ FM,
              const float* __restrict__ FJ,
              const int* __restrict__ IO, const int* __restrict__ IM,
              const int* __restrict__ IJ,
              const float* __restrict__ W, const float* __restrict__ b1,
              float* __restrict__ Y, float* __restrict__ stats, int N) {
  constexpr int K = 192, AS = 196;
  __shared__ float sA[32 * AS];
  __shared__ float ssum[64], ssq[64];
  const int tid = threadIdx.x;
  const int lane = tid & 31, wave = tid >> 5;
  const int lmod = lane & 15, lhalf = lane >> 4;
  const int mblk = blockIdx.x * 32;
  const int m0 = mblk + (wave >> 2) * 16;
  const int n0 = (wave & 3) * 16;

  if (tid < 64) { ssum[tid] = 0.0f; ssq[tid] = 0.0f; }
  for (int idx = tid; idx < 32 * 48; idx += 256) {
    int r = idx / 48, k = (idx % 48) * 4;   // float4 never straddles a segment
    int row = mblk + r; row = row < N ? row : N - 1;
    const float* tab;
    int off;
    if (k < 64)       { tab = FO + (long)IO[row] * 64; off = k; }
    else if (k < 128) { tab = FM + (long)IM[row] * 64; off = k - 64; }
    else              { tab = FJ + (long)IJ[row] * 64; off = k - 128; }
    *(float4*)&sA[r * AS + k] = *(const float4*)(tab + off);
  }
  __syncthreads();

  const int ar = (wave >> 2) * 16 + lmod;
  const int col = n0 + lmod;
  v8f c = {};
#pragma unroll
  for (int kk = 0; kk < K; kk += 4) {
    const int kb = kk + 2 * lhalf;
    v2f a = *(const v2f*)&sA[ar * AS + kb];
    v2f b;
    b.x = W[kb * 64 + col];
    b.y = W[(kb + 1) * 64 + col];
    c = __builtin_amdgcn_wmma_f32_16x16x4_f32(false, a, false, b, (short)0, c, false, false);
  }
  float bias = b1[col];
  float s = 0.0f, sq = 0.0f;
#pragma unroll
  for (int r = 0; r < 8; ++r) {
    int row = m0 + r + 8 * lhalf;
    if (row < N) {
      float v = c[r] + bias;
      Y[(long)row * 64 + col] = v;
      s += v; sq += v * v;
    }
  }
  atomicAdd(&ssum[col], s);
  atomicAdd(&ssq[col], sq);
  __syncthreads();
  if (tid < 64) {
    atomicAdd(&stats[tid], ssum[tid]);
    atomicAdd(&stats[64 + tid], ssq[tid]);
  }
}

// Scoring layer 2: BN1+ReLU staged, K=64, M=32. Block = 64 rows x 32 cols.
__global__ void __launch_bounds__(256)
score_h2_wmma(const float* __restrict__ H1, const float* __restrict__ stats1,
              const float* __restrict__ W, const float* __restrict__ b2,
              float* __restrict__ Y, float* __restrict__ stats2, int N) {
  constexpr int K = 64, AS = 68;
  __shared__ float sA[64 * AS];
  __shared__ float sW[32 * AS];
  __shared__ float ssum[32], ssq[32];
  const int tid = threadIdx.x;
  const int lane = tid & 31, wave = tid >> 5;
  const int lmod = lane & 15, lhalf = lane >> 4;
  const int mblk = blockIdx.x * 64;
  const int m0 = mblk + (wave >> 1) * 16;
  const int n0 = (wave & 1) * 16;
  const float* bnA = stats1 + 128;
  const float* bnB = stats1 + 192;

  if (tid < 32) { ssum[tid] = 0.0f; ssq[tid] = 0.0f; }
  for (int idx = tid; idx < 64 * 16; idx += 256) {
    int r = idx >> 4, k = (idx & 15) * 4;
    int row = mblk + r; row = row < N ? row : N - 1;
    float4 h = *(const float4*)(H1 + (long)row * 64 + k);
    float* d = &sA[r * AS + k];
    d[0] = fmaxf(bnA[k + 0] * h.x + bnB[k + 0], 0.0f);
    d[1] = fmaxf(bnA[k + 1] * h.y + bnB[k + 1], 0.0f);
    d[2] = fmaxf(bnA[k + 2] * h.z + bnB[k + 2], 0.0f);
    d[3] = fmaxf(bnA[k + 3] * h.w + bnB[k + 3], 0.0f);
  }
  for (int idx = tid; idx < K * 8; idx += 256) {
    int k = idx >> 3, c = (idx & 7) * 4;
    float4 w = *(const float4*)(W + k * 32 + c);
    sW[(c + 0) * AS + k] = w.x;
    sW[(c + 1) * AS + k] = w.y;
    sW[(c + 2) * AS + k] = w.z;
    sW[(c + 3) * AS + k] = w.w;
  }
  __syncthreads();

  const int ar = (wave >> 1) * 16 + lmod;
  const int col = n0 + lmod;
  v8f c = {};
#pragma unroll
  for (int kk = 0; kk < K; kk += 4) {
    const int kb = kk + 2 * lhalf;
    v2f a = *(const v2f*)&sA[ar * AS + kb];
    v2f b = *(const v2f*)&sW[col * AS + kb];
    c = __builtin_amdgcn_wmma_f32_16x16x4_f32(false, a, false, b, (short)0, c, false, false);
  }
  float bias = b2[col];
  float s = 0.0f, sq = 0.0f;
#pragma unroll
  for (int r = 0; r < 8; ++r) {
    int row = m0 + r + 8 * lhalf;
    if (row < N) {
      float v = c[r] + bias;
      Y[(long)row * 32 + col] = v;
      s += v; sq += v * v;
    }
  }
  atomicAdd(&ssum[col], s);
  atomicAdd(&ssq[col], sq);
  __syncthreads();
  if (tid < 32) {
    atomicAdd(&stats2[tid], ssum[tid]);
    atomicAdd(&stats2[64 + tid], ssq[tid]);
  }
}

// Final: out[p] = ReLU(bnA2*h2+bnB2) . Ws3 + bs3
__global__ void score_final_kernel(const float* __restrict__ H2, const float* __restrict__ stats2,
                                   const float* __restrict__ W3, const float* __restrict__ b3,
                                   float* __restrict__ out, int N) {
  int p = blockIdx.x * blockDim.x + threadIdx.x;
  if (p >= N) return;
  const float* bnA = stats2 + 128;
  const float* bnB = stats2 + 192;
  float acc = b3[0];
#pragma unroll
  for (int j = 0; j < 32; ++j) {
    float z = bnA[j] * H2[(long)p * 32 + j] + bnB[j];
    acc += fmaxf(z, 0.0f) * W3[j];
  }
  out[p] = acc;
}

// ------------------------------------------------------------------- launch
extern "C" void kernel_launch(void* const* d_in, const int* in_sizes, int n_in,
                              void* d_out, int out_size, void* d_ws, size_t ws_size,
                              hipStream_t stream) {
  const float* x_op   = (const float*)d_in[0];
  const float* x_ma   = (const float*)d_in[1];
  const float* x_jb   = (const float*)d_in[2];
  const float* Weo    = (const float*)d_in[3];
  const float* Wem    = (const float*)d_in[4];
  const float* Wej    = (const float*)d_in[5];
  const float* gW1_0  = (const float*)d_in[6];   // [4,32,64]
  const float* gW1_12 = (const float*)d_in[7];   // [2,4,64,64]
  const float* gb1    = (const float*)d_in[8];   // [3,4,64]
  const float* gg1    = (const float*)d_in[9];
  const float* gbe1   = (const float*)d_in[10];
  const float* gW2    = (const float*)d_in[11];  // [3,4,64,64]
  const float* gb2    = (const float*)d_in[12];
  const float* geps   = (const float*)d_in[13];  // [3,4]
  const float* Ws1    = (const float*)d_in[14];  // [192,64]
  const float* bs1    = (const float*)d_in[15];
  const float* gs1    = (const float*)d_in[16];
  const float* bes1   = (const float*)d_in[17];
  const float* Ws2    = (const float*)d_in[18];  // [64,32]
  const float* bs2    = (const float*)d_in[19];
  const float* gs2    = (const float*)d_in[20];
  const float* bes2   = (const float*)d_in[21];
  const float* Ws3    = (const float*)d_in[22];  // [32,1]
  const float* bs3    = (const float*)d_in[23];
  const int* e_om_src = (const int*)d_in[24];
  const int* e_om_dst = (const int*)d_in[25];
  const int* e_mo_src = (const int*)d_in[26];
  const int* e_mo_dst = (const int*)d_in[27];
  const int* e_jo_src = (const int*)d_in[28];
  const int* e_jo_dst = (const int*)d_in[29];
  const int* e_oj_src = (const int*)d_in[30];
  const int* e_oj_dst = (const int*)d_in[31];
  const int* vp_op    = (const int*)d_in[32];
  const int* vp_ma    = (const int*)d_in[33];
  const int* vp_jb    = (const int*)d_in[34];

  char* wp = (char*)d_ws;
  auto alloc = [&](size_t nfloats) -> float* {
    float* r = (float*)wp;
    wp += ((nfloats * sizeof(float) + 255) / 256) * 256;
    return r;
  };
  float* opA  = alloc((size_t)N_OP * 64);
  float* opB  = alloc((size_t)N_OP * 64);
  float* maA  = alloc((size_t)N_MACH * 64);
  float* maB  = alloc((size_t)N_MACH * 64);
  float* jbA  = alloc((size_t)N_JOB * 64);
  float* jbB  = alloc((size_t)N_JOB * 64);
  float* agg  = alloc((size_t)N_OP * 64);
  float* h1   = alloc((size_t)N_OP * 64);
  float* s1   = alloc((size_t)N_PAIRS * 64);
  float* s2   = alloc((size_t)N_PAIRS * 32);
  float* stats  = alloc(256);
  float* stats2 = alloc(256);

  auto zero = [&](float* p, long n) {
    zero_kernel<<<(unsigned)((n + 255) / 256), 256, 0, stream>>>(p, n);
  };
  auto scat = [&](const float* src, const int* si, const int* di, float* ag, int E, int F) {
    long tot = (long)E * (F / 4);
    scatter_add4_kernel<<<(unsigned)((tot + 255) / 256), 256, 0, stream>>>(src, si, di, ag, E, F);
  };
  auto run_gin = [&](const float* xin, int Nn, int Kin, const float* aggp,
                     int l, int e, const float* resid, int accum, float* yout) {
    const float* W1 = (l == 0) ? (gW1_0 + (size_t)e * 32 * 64)
                               : (gW1_12 + (size_t)((l - 1) * 4 + e) * 64 * 64);
    size_t pe = (size_t)(l * 4 + e);
    const float* b1  = gb1  + pe * 64;
    const float* g1  = gg1  + pe * 64;
    const float* be1 = gbe1 + pe * 64;
    const float* W2  = gW2  + pe * 64 * 64;
    const float* b2  = gb2  + pe * 64;
    const float* ep  = geps + pe;
    zero(stats, 128);
    unsigned grid = (unsigned)((Nn + 31) / 32);
    if (Kin == 32)
      gin_h1_wmma<32><<<grid, 256, 0, stream>>>(xin, aggp, ep, W1, b1, h1, stats, Nn);
    else
      gin_h1_wmma<64><<<grid, 256, 0, stream>>>(xin, aggp, ep, W1, b1, h1, stats, Nn);
    bn_finalize_kernel<<<1, 64, 0, stream>>>(stats, g1, be1, 1.0f / (float)Nn, 64);
    gin_h2_wmma<<<grid, 256, 0, stream>>>(h1, stats, W2, b2, resid, yout, Nn, accum);
  };

  // ---- encoders
  encode_kernel<<<(N_OP + 255) / 256, 256, 0, stream>>>(x_op, Weo, opA, N_OP, 8);
  encode_kernel<<<(N_MACH + 255) / 256, 256, 0, stream>>>(x_ma, Wem, maA, N_MACH, 4);
  encode_kernel<<<(N_JOB + 255) / 256, 256, 0, stream>>>(x_jb, Wej, jbA, N_JOB, 4);

  float *cur_op = opA, *new_op = opB;
  float *cur_ma = maA, *new_ma = maB;
  float *cur_jb = jbA, *new_jb = jbB;

  // ---- 3 GIN layers
  for (int l = 0; l < 3; ++l) {
    int Kin = (l == 0) ? 32 : 64;
    // edge type 0: op -> mach
    zero(agg, (long)N_MACH * Kin);
    scat(cur_op, e_om_src, e_om_dst, agg, E_OM, Kin);
    run_gin(cur_ma, N_MACH, Kin, agg, l, 0, l ? cur_ma : nullptr, 0, new_ma);
    // edge type 1: mach -> op
    zero(agg, (long)N_OP * Kin);
    scat(cur_ma, e_mo_src, e_mo_dst, agg, E_OM, Kin);
    run_gin(cur_op, N_OP, Kin, agg, l, 1, l ? cur_op : nullptr, 0, new_op);
    // edge type 2: job -> op (hetero-sum accumulate into new_op)
    zero(agg, (long)N_OP * Kin);
    scat(cur_jb, e_jo_src, e_jo_dst, agg, E_JO, Kin);
    run_gin(cur_op, N_OP, Kin, agg, l, 2, nullptr, 1, new_op);
    // edge type 3: op -> job
    zero(agg, (long)N_JOB * Kin);
    scat(cur_op, e_oj_src, e_oj_dst, agg, E_JO, Kin);
    run_gin(cur_jb, N_JOB, Kin, agg, l, 3, l ? cur_jb : nullptr, 0, new_jb);

    float* t;
    t = cur_op; cur_op = new_op; new_op = t;
    t = cur_ma; cur_ma = new_ma; new_ma = t;
    t = cur_jb; cur_jb = new_jb; new_jb = t;
  }

  // ---- scoring MLP on 500K pairs (gather fused into GEMM A staging)
  zero(stats, 128);
  score_h1_wmma<<<(N_PAIRS + 31) / 32, 256, 0, stream>>>(
      cur_op, cur_ma, cur_jb, vp_op, vp_ma, vp_jb, Ws1, bs1, s1, stats, N_PAIRS);
  bn_finalize_kernel<<<1, 64, 0, stream>>>(stats, gs1, bes1, 1.0f / (float)N_PAIRS, 64);

  zero(stats2, 128);
  score_h2_wmma<<<(N_PAIRS + 63) / 64, 256, 0, stream>>>(
      s1, stats, Ws2, bs2, s2, stats2, N_PAIRS);
  bn_finalize_kernel<<<1, 64, 0, stream>>>(stats2, gs2, bes2, 1.0f / (float)N_PAIRS, 32);

  score_final_kernel<<<(N_PAIRS + 255) / 256, 256, 0, stream>>>(
      s2, stats2, Ws3, bs3, (float*)d_out, N_PAIRS);
}